// BallCrossEntropy_35064113005059
// MI455X (gfx1250) — compile-verified
//
#include <hip/hip_runtime.h>
#include <stdint.h>

// ---------------------------------------------------------------------------
// BallCrossEntropy: mean NLL over all (B,H,W) positions of NCHW fp32 logits.
//   B=4, C=256, H=288, W=512.  Memory-bound streaming reduction (~604 MB).
// Strategy: one block per (b,h) row; 128 threads x 4 pixels (float4 along W).
// Channels streamed in 8-channel chunks, double-buffered into LDS with
// gfx1250 GLOBAL_LOAD_ASYNC_TO_LDS_B128 (ASYNCcnt / s_wait_asynccnt).
// Online base-2 softmax: one v_exp_f32 per element.
// Deterministic two-stage reduction (no float atomics).
// ---------------------------------------------------------------------------

namespace {
constexpr int   B_ = 4, C_ = 256, H_ = 288, W_ = 512;
constexpr long long HW_ = (long long)H_ * W_;      // 147456
constexpr int   TPB  = 128;                        // 4 x wave32; one row of W
constexpr int   PIX  = 4;                          // pixels per thread
constexpr int   CH   = 8;                          // channels per LDS chunk
constexpr int   NCHUNK = C_ / CH;                  // 32
constexpr int   NBLK = B_ * H_;                    // 1152 partial sums
constexpr float LOG2E_F = 1.44269504088896340736f;
constexpr float LN2_F   = 0.69314718055994530942f;
constexpr float INV_N   = 1.0f / (float)((long long)B_ * H_ * W_);
}

// ---- device math helpers (guard amdgcn builtins from the host pass) -------
__device__ __forceinline__ float fast_exp2(float x) {
#if defined(__HIP_DEVICE_COMPILE__)
  return __builtin_amdgcn_exp2f(x);        // v_exp_f32 (base-2)
#else
  return exp2f(x);
#endif
}
__device__ __forceinline__ float fast_log2(float x) {
#if defined(__HIP_DEVICE_COMPILE__)
  return __builtin_amdgcn_logf(x);         // v_log_f32 (base-2)
#else
  return log2f(x);
#endif
}

// Online softmax accumulator update in base-2 domain (1 transcendental/elem).
__device__ __forceinline__ void online_update(float& m, float& s, float x) {
  float y  = x * LOG2E_F;
  float mo = m;
  m = fmaxf(mo, y);
  float e = fast_exp2(-fabsf(y - mo));     // exp2(min(y,mo) - max(y,mo))
  // if y > mo: s = s*e + 1  (rescale old sum, new term is 2^0)
  // else:      s = s + e    (new term relative to existing max)
  s = (y > mo) ? fmaf(s, e, 1.0f) : (s + e);
}

// ---- gfx1250 async global->LDS path, with safe fallback --------------------
#if defined(__HIP_DEVICE_COMPILE__) && \
    __has_builtin(__builtin_amdgcn_global_load_async_to_lds_b128)
#define HAVE_ASYNC_LDS 1
typedef int v4i_t __attribute__((ext_vector_type(4)));
typedef __attribute__((address_space(1))) v4i_t GlbV4;   // global int4*
typedef __attribute__((address_space(3))) v4i_t LdsV4;   // LDS int4*
__device__ __forceinline__ void async_ld_b128(const float* g, float* l) {
  __builtin_amdgcn_global_load_async_to_lds_b128((GlbV4*)g, (LdsV4*)l, 0, 0);
}
__device__ __forceinline__ void wait_asynccnt_8() {
#if __has_builtin(__builtin_amdgcn_s_wait_asynccnt)
  __builtin_amdgcn_s_wait_asynccnt(CH);
#else
  asm volatile("s_wait_asynccnt %0" ::"i"(CH) : "memory");
#endif
}
__device__ __forceinline__ void wait_asynccnt_0() {
#if __has_builtin(__builtin_amdgcn_s_wait_asynccnt)
  __builtin_amdgcn_s_wait_asynccnt(0);
#else
  asm volatile("s_wait_asynccnt 0" ::: "memory");
#endif
}
#endif

__global__ __launch_bounds__(TPB) void ce_row_kernel(
    const float* __restrict__ logits, const int* __restrict__ target,
    float* __restrict__ partial) {
  const int blk = blockIdx.x;
  const int b   = blk / H_;
  const int h   = blk % H_;
  const int tid = threadIdx.x;
  const int w0  = tid * PIX;
  const size_t rowbase = ((size_t)b * C_) * (size_t)HW_ + (size_t)h * W_;

  float m[PIX], s[PIX];
#pragma unroll
  for (int p = 0; p < PIX; ++p) { m[p] = -INFINITY; s[p] = 0.0f; }

#if defined(HAVE_ASYNC_LDS)
  __shared__ alignas(16) float buf[2][CH * W_];   // 2 x 16KB double buffer

  // Prime the pipeline: chunk 0 -> buf[0].
  {
    const float* g = logits + rowbase + w0;
#pragma unroll
    for (int j = 0; j < CH; ++j)
      async_ld_b128(g + (size_t)j * HW_, &buf[0][j * W_ + w0]);
  }

  for (int ch = 0; ch < NCHUNK; ++ch) {
    const int pb = ch & 1;
    if (ch + 1 < NCHUNK) {
      // Issue next chunk into the other buffer, then wait for current chunk.
      const float* g = logits + rowbase + (size_t)(ch + 1) * CH * HW_ + w0;
#pragma unroll
      for (int j = 0; j < CH; ++j)
        async_ld_b128(g + (size_t)j * HW_, &buf[pb ^ 1][j * W_ + w0]);
      wait_asynccnt_8();   // <= 8 outstanding => chunk `ch` complete (in-order)
    } else {
      wait_asynccnt_0();
    }
    __syncthreads();       // all waves' async writes for chunk `ch` visible

#pragma unroll
    for (int j = 0; j < CH; ++j) {
      float4 v = *reinterpret_cast<const float4*>(&buf[pb][j * W_ + w0]);
      online_update(m[0], s[0], v.x);
      online_update(m[1], s[1], v.y);
      online_update(m[2], s[2], v.z);
      online_update(m[3], s[3], v.w);
    }
    __syncthreads();       // buffer `pb` free for reuse next iteration
  }
#else
  // Fallback: direct coalesced b128 streaming (no LDS staging).
#pragma unroll 8
  for (int c = 0; c < C_; ++c) {
    float4 v = *reinterpret_cast<const float4*>(
        logits + rowbase + (size_t)c * HW_ + w0);
    online_update(m[0], s[0], v.x);
    online_update(m[1], s[1], v.y);
    online_update(m[2], s[2], v.z);
    online_update(m[3], s[3], v.w);
  }
#endif

  // Per-pixel loss: ln2*(m + log2 s) - x[target]
  float loss = 0.0f;
  const int tbase = (b * H_ + h) * W_ + w0;
#pragma unroll
  for (int p = 0; p < PIX; ++p) {
    const int w = w0 + p;
    const int t = target[tbase + p];
    const float xt  = logits[rowbase + (size_t)t * HW_ + w];
    const float lse = (m[p] + fast_log2(s[p])) * LN2_F;
    loss += lse - xt;
  }

  // Deterministic block reduction.
  __shared__ float red[TPB];
  red[tid] = loss;
  __syncthreads();
#pragma unroll
  for (int off = TPB / 2; off > 0; off >>= 1) {
    if (tid < off) red[tid] += red[tid + off];
    __syncthreads();
  }
  if (tid == 0) partial[blk] = red[0];
}

__global__ __launch_bounds__(256) void ce_finalize_kernel(
    const float* __restrict__ partial, float* __restrict__ out) {
  __shared__ float red[256];
  float s = 0.0f;
  for (int i = threadIdx.x; i < NBLK; i += 256) s += partial[i];
  red[threadIdx.x] = s;
  __syncthreads();
#pragma unroll
  for (int off = 128; off > 0; off >>= 1) {
    if (threadIdx.x < off) red[threadIdx.x] += red[threadIdx.x + off];
    __syncthreads();
  }
  if (threadIdx.x == 0) out[0] = red[0] * INV_N;
}

extern "C" void kernel_launch(void* const* d_in, const int* in_sizes, int n_in,
                              void* d_out, int out_size, void* d_ws,
                              size_t ws_size, hipStream_t stream) {
  const float* logits = (const float*)d_in[0];   // [B,C,H,W] fp32
  const int*   target = (const int*)d_in[1];     // [B,H,W] int32
  float*       out    = (float*)d_out;           // scalar
  float*       part   = (float*)d_ws;            // NBLK floats of scratch

  ce_row_kernel<<<NBLK, TPB, 0, stream>>>(logits, target, part);
  ce_finalize_kernel<<<1, 256, 0, stream>>>(part, out);
}